// EnsembleE2EModule_19756849562150
// MI455X (gfx1250) — compile-verified
//
#include <hip/hip_runtime.h>
#include <hip/hip_bf16.h>
#include <math.h>

// Problem sizes (fixed by the reference)
#define BATCH 4096
#define DIM   784
#define NCLS  10
#define NEXP  1024
#define KNN   8

typedef __attribute__((ext_vector_type(2))) float v2f;
typedef __attribute__((ext_vector_type(8))) float v8f;

#define NEG_HUGE (-3.402823466e38f)

// ---------------------------------------------------------------------------
// K1: reciprocal row norms  rnorm[b] = 1 / max(||x_b||, 1e-12)
// one wave (32 lanes) per row, 8 rows per 256-thread block
// ---------------------------------------------------------------------------
__global__ __launch_bounds__(256) void rnorm_kernel(const float* __restrict__ x,
                                                    float* __restrict__ rnorm) {
    const int wave = threadIdx.x >> 5;
    const int lane = threadIdx.x & 31;
    const int b = blockIdx.x * 8 + wave;
    const float* xr = x + (size_t)b * DIM;
    float ss = 0.f;
    for (int d = lane; d < DIM; d += 32) {
        float v = xr[d];
        ss = fmaf(v, v, ss);
    }
    #pragma unroll
    for (int off = 16; off >= 1; off >>= 1) ss += __shfl_xor(ss, off, 32);
    if (lane == 0) rnorm[b] = 1.0f / fmaxf(sqrtf(ss), 1e-12f);
}

// ---------------------------------------------------------------------------
// K2: cos[b,e] = rnorm[b] * (x[b,:] . keys[e,:])   via V_WMMA_F32_16X16X4_F32
// Block tile 128(M) x 64(N), K-chunks of 64 in LDS. 8 waves: wave w owns rows
// 16w..16w+15 and all 64 cols (4 x 16x16 accumulators).
// ---------------------------------------------------------------------------
#define BM 128
#define BN 64
#define BKC 64
#define LDA 66  // padded LDS stride (dwords), even -> 8B-aligned b64 loads
#define LDB 66

__global__ __launch_bounds__(256) void cos_gemm_kernel(const float* __restrict__ x,
                                                       const float* __restrict__ keys,
                                                       const float* __restrict__ rnorm,
                                                       float* __restrict__ cosm) {
    __shared__ float As[BM * LDA];
    __shared__ float Bs[BN * LDB];

    const int tid  = threadIdx.x;
    const int wave = tid >> 5;
    const int lane = tid & 31;
    const int half = lane >> 4;   // 0: lanes 0-15, 1: lanes 16-31
    const int lid  = lane & 15;
    const int rowBase = blockIdx.y * BM;
    const int eBase   = blockIdx.x * BN;

    v8f acc[4];
    #pragma unroll
    for (int j = 0; j < 4; ++j) acc[j] = (v8f){0.f,0.f,0.f,0.f,0.f,0.f,0.f,0.f};

    const int c4 = tid & 15;       // which float4 within the 64-wide K chunk
    const int r0 = tid >> 4;       // 0..15

    for (int kk = 0; kk < DIM; kk += BKC) {
        const int col = kk + (c4 << 2);
        const bool inb = (col < DIM);  // DIM % 4 == 0, col multiple of 4
        // stage A: 128 x 64 fp32 (x rows)
        #pragma unroll
        for (int p = 0; p < 8; ++p) {
            const int r = r0 + (p << 4);
            float4 v = make_float4(0.f, 0.f, 0.f, 0.f);
            if (inb) v = *(const float4*)(x + (size_t)(rowBase + r) * DIM + col);
            float* dst = &As[r * LDA + (c4 << 2)];
            dst[0] = v.x; dst[1] = v.y; dst[2] = v.z; dst[3] = v.w;
        }
        // stage B: 64 x 64 fp32 (keys rows; B[k][n] = keys[eBase+n][kk+k])
        #pragma unroll
        for (int p = 0; p < 4; ++p) {
            const int r = r0 + (p << 4);
            float4 v = make_float4(0.f, 0.f, 0.f, 0.f);
            if (inb) v = *(const float4*)(keys + (size_t)(eBase + r) * DIM + col);
            float* dst = &Bs[r * LDB + (c4 << 2)];
            dst[0] = v.x; dst[1] = v.y; dst[2] = v.z; dst[3] = v.w;
        }
        __syncthreads();

        const float* arow = &As[(16 * wave + lid) * LDA];
        #pragma unroll
        for (int ks = 0; ks < 16; ++ks) {
            const int k0 = 4 * ks + 2 * half;           // even -> 8B aligned
            v2f a = *(const v2f*)(arow + k0);           // A[M=lid][k0], [k0+1]
            #pragma unroll
            for (int j = 0; j < 4; ++j) {
                v2f bb = *(const v2f*)(&Bs[(16 * j + lid) * LDB + k0]); // B[N=lid][k0..k0+1]
                acc[j] = __builtin_amdgcn_wmma_f32_16x16x4_f32(
                    false, a, false, bb, (short)0, acc[j], false, false);
            }
        }
        __syncthreads();
    }

    // epilogue: scale by rnorm and store. C/D: VGPR v -> M = v + 8*half, N = lid
    #pragma unroll
    for (int j = 0; j < 4; ++j) {
        const int e = eBase + 16 * j + lid;
        #pragma unroll
        for (int v = 0; v < 8; ++v) {
            const int brow = rowBase + 16 * wave + v + 8 * half;
            cosm[(size_t)brow * NEXP + e] = acc[j][v] * rnorm[brow];
        }
    }
}

// ---------------------------------------------------------------------------
// K3: top-8 per row (descending sims + ascending-sorted indices)
// one wave per row, row staged in LDS; 8x (argmax + mask)
// ---------------------------------------------------------------------------
__global__ __launch_bounds__(256) void topk_kernel(const float* __restrict__ cosm,
                                                   float* __restrict__ sims_out,
                                                   int* __restrict__ gidx_out) {
    __shared__ float rowbuf[8][NEXP];
    const int wave = threadIdx.x >> 5;
    const int lane = threadIdx.x & 31;
    const int b = blockIdx.x * 8 + wave;
    float* row = rowbuf[wave];
    const float* src = cosm + (size_t)b * NEXP;
    for (int i = lane; i < NEXP; i += 32) row[i] = src[i];
    __builtin_amdgcn_wave_barrier();

    float simv[KNN];
    int   simi[KNN];
    #pragma unroll
    for (int t = 0; t < KNN; ++t) {
        float bestv = NEG_HUGE;
        int   besti = 0x7fffffff;
        for (int i = lane; i < NEXP; i += 32) {
            float v = row[i];
            if (v > bestv || (v == bestv && i < besti)) { bestv = v; besti = i; }
        }
        #pragma unroll
        for (int off = 16; off >= 1; off >>= 1) {
            float ov = __shfl_xor(bestv, off, 32);
            int   oi = __shfl_xor(besti, off, 32);
            if (ov > bestv || (ov == bestv && oi < besti)) { bestv = ov; besti = oi; }
        }
        simv[t] = bestv;
        simi[t] = besti;
        if (lane == 0) row[besti] = NEG_HUGE;  // DS ops in-order within wave
        __builtin_amdgcn_wave_barrier();
    }

    // sort the 8 chosen indices ascending (sims stay similarity-sorted)
    int s[KNN];
    #pragma unroll
    for (int t = 0; t < KNN; ++t) s[t] = simi[t];
#define CSWAP(a, b) { int lo = min(s[a], s[b]); int hi = max(s[a], s[b]); s[a] = lo; s[b] = hi; }
    CSWAP(0,1) CSWAP(2,3) CSWAP(4,5) CSWAP(6,7)
    CSWAP(0,2) CSWAP(1,3) CSWAP(4,6) CSWAP(5,7)
    CSWAP(1,2) CSWAP(5,6) CSWAP(0,4) CSWAP(3,7)
    CSWAP(1,5) CSWAP(2,6)
    CSWAP(1,4) CSWAP(3,6)
    CSWAP(2,4) CSWAP(3,5)
    CSWAP(3,4)
#undef CSWAP
    if (lane == 0) {
        #pragma unroll
        for (int t = 0; t < KNN; ++t) {
            sims_out[(size_t)b * KNN + t] = simv[t];
            gidx_out[(size_t)b * KNN + t] = s[t];
        }
    }
}

// ---------------------------------------------------------------------------
// K4: gathered-expert GEMVs + tanh + sim-weighted average
// block = one sample b; wave k handles expert gidx[b,k] (Wm L2-resident: 32MB)
// ---------------------------------------------------------------------------
__global__ __launch_bounds__(256) void ensemble_kernel(const float* __restrict__ x,
                                                       const float* __restrict__ Wm,
                                                       const float* __restrict__ bm,
                                                       const float* __restrict__ sims,
                                                       const int* __restrict__ gidx,
                                                       float* __restrict__ ensemble) {
    __shared__ float eo[KNN][NCLS];
    __shared__ float ew[KNN];
    const int wave = threadIdx.x >> 5;
    const int lane = threadIdx.x & 31;
    const int b = blockIdx.x;
    const int e = gidx[(size_t)b * KNN + wave];
    const float wk = sims[(size_t)b * KNN + wave];

    const float* xr = x + (size_t)b * DIM;
    const float* Wr = Wm + (size_t)e * NCLS * DIM;

    float acc[NCLS];
    #pragma unroll
    for (int c = 0; c < NCLS; ++c) acc[c] = 0.f;

    for (int d = lane; d < DIM; d += 32) {
        const float xv = xr[d];
        #pragma unroll
        for (int c = 0; c < NCLS; ++c) acc[c] = fmaf(xv, Wr[c * DIM + d], acc[c]);
    }
    #pragma unroll
    for (int c = 0; c < NCLS; ++c) {
        float v = acc[c];
        #pragma unroll
        for (int off = 16; off >= 1; off >>= 1) v += __shfl_xor(v, off, 32);
        if (lane == 0) eo[wave][c] = tanhf((v + bm[e * NCLS + c]) * 0.1f) * 10.0f;
    }
    if (lane == 0) ew[wave] = wk;
    __syncthreads();

    if (threadIdx.x < NCLS) {
        float num = 0.f, den = 0.f;
        #pragma unroll
        for (int k = 0; k < KNN; ++k) {
            num = fmaf(ew[k], eo[k][threadIdx.x], num);
            den += ew[k];
        }
        ensemble[(size_t)b * NCLS + threadIdx.x] = num / den;
    }
}

// ---------------------------------------------------------------------------
// K5: vanilla (log_softmax) + tanh classifiers, one wave per row
// ---------------------------------------------------------------------------
__global__ __launch_bounds__(256) void classifier_kernel(const float* __restrict__ x,
                                                         const float* __restrict__ Wv,
                                                         const float* __restrict__ bv,
                                                         const float* __restrict__ Wt,
                                                         const float* __restrict__ bt,
                                                         float* __restrict__ tanh_out,
                                                         float* __restrict__ vanilla_out) {
    const int wave = threadIdx.x >> 5;
    const int lane = threadIdx.x & 31;
    const int b = blockIdx.x * 8 + wave;
    const float* xr = x + (size_t)b * DIM;

    float av[NCLS], at[NCLS];
    #pragma unroll
    for (int c = 0; c < NCLS; ++c) { av[c] = 0.f; at[c] = 0.f; }

    for (int d = lane; d < DIM; d += 32) {
        const float xv = xr[d];
        #pragma unroll
        for (int c = 0; c < NCLS; ++c) {
            av[c] = fmaf(xv, Wv[c * DIM + d], av[c]);
            at[c] = fmaf(xv, Wt[c * DIM + d], at[c]);
        }
    }
    #pragma unroll
    for (int c = 0; c < NCLS; ++c) {
        #pragma unroll
        for (int off = 16; off >= 1; off >>= 1) {
            av[c] += __shfl_xor(av[c], off, 32);
            at[c] += __shfl_xor(at[c], off, 32);
        }
    }
    if (lane == 0) {
        float lv[NCLS];
        float m = NEG_HUGE;
        #pragma unroll
        for (int c = 0; c < NCLS; ++c) { lv[c] = av[c] + bv[c]; m = fmaxf(m, lv[c]); }
        float ssum = 0.f;
        #pragma unroll
        for (int c = 0; c < NCLS; ++c) ssum += expf(lv[c] - m);
        const float lse = logf(ssum);
        #pragma unroll
        for (int c = 0; c < NCLS; ++c)
            vanilla_out[(size_t)b * NCLS + c] = lv[c] - m - lse;
        #pragma unroll
        for (int c = 0; c < NCLS; ++c)
            tanh_out[(size_t)b * NCLS + c] = tanhf((at[c] + bt[c]) * 0.1f) * 10.0f;
    }
}

// ---------------------------------------------------------------------------
extern "C" void kernel_launch(void* const* d_in, const int* in_sizes, int n_in,
                              void* d_out, int out_size, void* d_ws, size_t ws_size,
                              hipStream_t stream) {
    const float* x    = (const float*)d_in[0];
    const float* keys = (const float*)d_in[1];
    const float* Wm   = (const float*)d_in[2];
    const float* bm   = (const float*)d_in[3];
    const float* Wv   = (const float*)d_in[4];
    const float* bv   = (const float*)d_in[5];
    const float* Wt   = (const float*)d_in[6];
    const float* bt   = (const float*)d_in[7];

    float* out      = (float*)d_out;
    float* ensemble = out;                         // [B, C]
    float* tanh_out = out + (size_t)BATCH * NCLS;  // [B, C]
    float* vanilla  = out + (size_t)2 * BATCH * NCLS;

    float* ws    = (float*)d_ws;
    float* rnorm = ws;                               // BATCH
    float* cosm  = ws + BATCH;                       // BATCH*NEXP
    float* sims  = cosm + (size_t)BATCH * NEXP;      // BATCH*KNN
    int*   gidx  = (int*)(sims + (size_t)BATCH * KNN);

    rnorm_kernel<<<BATCH / 8, 256, 0, stream>>>(x, rnorm);

    dim3 g2(NEXP / BN, BATCH / BM);  // (16, 32)
    cos_gemm_kernel<<<g2, 256, 0, stream>>>(x, keys, rnorm, cosm);

    topk_kernel<<<BATCH / 8, 256, 0, stream>>>(cosm, sims, gidx);

    ensemble_kernel<<<BATCH, 256, 0, stream>>>(x, Wm, bm, sims, gidx, ensemble);

    classifier_kernel<<<BATCH / 8, 256, 0, stream>>>(x, Wv, bv, Wt, bt, tanh_out, vanilla);
}